// DrugGraph_21809843929588
// MI455X (gfx1250) — compile-verified
//
#include <hip/hip_runtime.h>
#include <hip/hip_bf16.h>

#define DIM 128
#define NEG_SLOPE 0.1f

typedef float v2f __attribute__((ext_vector_type(2)));
typedef float v8f __attribute__((ext_vector_type(8)));

// ---------------------------------------------------------------------------
// Edge preprocessing (layer-invariant)
// ---------------------------------------------------------------------------
__global__ void k_init_deg(float* __restrict__ deg, int N) {
    int i = blockIdx.x * blockDim.x + threadIdx.x;
    if (i < N) deg[i] = 1.0f;  // self-loop weight
}

__global__ void k_w_deg(const float* __restrict__ attr, const int* __restrict__ etype,
                        const float* __restrict__ scale, const int* __restrict__ dst,
                        float* __restrict__ w, float* __restrict__ deg, int E) {
    int e = blockIdx.x * blockDim.x + threadIdx.x;
    if (e >= E) return;
    float wv = scale[etype[e]] * attr[e];
    w[e] = wv;
    atomicAdd(&deg[dst[e]], wv);
}

__global__ void k_dinv(float* __restrict__ deg_to_dinv, float* __restrict__ selfnorm, int N) {
    int i = blockIdx.x * blockDim.x + threadIdx.x;
    if (i >= N) return;
    float d = deg_to_dinv[i];           // >= 1.0 always (self loop)
    float r = rsqrtf(d);
    deg_to_dinv[i] = r;                 // in-place: deg buffer becomes dinv
    selfnorm[i]    = r * r;             // self-loop norm = 1/deg
}

__global__ void k_norm(const float* __restrict__ w, const float* __restrict__ dinv,
                       const int* __restrict__ src, const int* __restrict__ dst,
                       float* __restrict__ norm, int E) {
    int e = blockIdx.x * blockDim.x + threadIdx.x;
    if (e >= E) return;
    norm[e] = dinv[src[e]] * w[e] * dinv[dst[e]];
}

// ---------------------------------------------------------------------------
// GEMM: C[N x 128] = leaky(A)[N x 128] @ W[128 x 128] using V_WMMA_F32_16X16X4_F32
// One wave per 16x16 C tile. N % 16 == 0 and DIM == 128 exactly -> no
// divergence, EXEC is all-ones as WMMA requires.
//   blockDim = 256 (8 waves); block b -> row tile b; wave w -> col tile w.
// slope == 1.0f makes the fused leaky-ReLU an identity (layer 1).
// ---------------------------------------------------------------------------
__global__ __launch_bounds__(256) void k_gemm_wmma(const float* __restrict__ A,
                                                   const float* __restrict__ W,
                                                   float* __restrict__ C,
                                                   float slope) {
    const int wave = threadIdx.x >> 5;   // 0..7  -> column tile
    const int lane = threadIdx.x & 31;
    const int half = lane >> 4;          // 0 or 1 (selects K+2 per ISA layout)
    const int l16  = lane & 15;
    const int row0 = blockIdx.x * 16;
    const int col0 = wave * 16;

    const float* arow = A + (size_t)(row0 + l16) * DIM;   // A row for this lane (M = l16)
    v8f acc = {};

    #pragma unroll 4
    for (int k0 = 0; k0 < DIM; k0 += 4) {
        const int ka = k0 + 2 * half;
        // A frag (16x4 f32): VGPR0 = K=ka, VGPR1 = K=ka+1; leaky fused on load.
        float ax = arow[ka];
        float ay = arow[ka + 1];
        ax = ax > 0.0f ? ax : slope * ax;
        ay = ay > 0.0f ? ay : slope * ay;
        v2f a = { ax, ay };
        // B frag (4x16 f32): row K striped across lanes (N = col0+l16),
        // lane half selects K+2, consecutive VGPRs consecutive K.
        const float* wp = W + (size_t)ka * DIM + col0 + l16;
        v2f b = { wp[0], wp[DIM] };
        acc = __builtin_amdgcn_wmma_f32_16x16x4_f32(
            /*neg_a=*/false, a, /*neg_b=*/false, b,
            /*c_mod=*/(short)0, acc, /*reuse_a=*/false, /*reuse_b=*/false);
    }

    // D layout: VGPR r -> M = r + 8*half, N = col0 + l16
    float* cout = C + (size_t)(row0 + 8 * half) * DIM + col0 + l16;
    #pragma unroll
    for (int r = 0; r < 8; ++r) cout[(size_t)r * DIM] = acc[r];
}

// ---------------------------------------------------------------------------
// out[i,f] = selfnorm[i] * h[i,f] + bias[f]   (fully overwrites out)
// ---------------------------------------------------------------------------
__global__ void k_self_bias(const float* __restrict__ h, const float* __restrict__ selfnorm,
                            const float* __restrict__ bias, float* __restrict__ out, int total) {
    int idx = blockIdx.x * blockDim.x + threadIdx.x;
    if (idx >= total) return;
    int i = idx >> 7;          // / DIM
    int f = idx & (DIM - 1);
    out[idx] = selfnorm[i] * h[idx] + bias[f];
}

// ---------------------------------------------------------------------------
// Edge scatter: one wave per edge, float4 per lane (128 feats / 32 lanes).
// out[dst] += norm[e] * h[src]   via global_atomic_add_f32
// ---------------------------------------------------------------------------
__global__ __launch_bounds__(256) void k_scatter(const float* __restrict__ h,
                                                 const float* __restrict__ norm,
                                                 const int* __restrict__ src,
                                                 const int* __restrict__ dst,
                                                 float* __restrict__ out, int E) {
    int e = blockIdx.x * (blockDim.x >> 5) + (threadIdx.x >> 5);
    if (e >= E) return;
    const int lane = threadIdx.x & 31;
    const float n = norm[e];
    const int s = src[e];
    const int d = dst[e];
    const float4* hp = (const float4*)(h + (size_t)s * DIM);
    float4 v = hp[lane];                      // global_load_b128
    float* op = out + (size_t)d * DIM + lane * 4;
    atomicAdd(op + 0, n * v.x);
    atomicAdd(op + 1, n * v.y);
    atomicAdd(op + 2, n * v.z);
    atomicAdd(op + 3, n * v.w);
}

// ---------------------------------------------------------------------------
extern "C" void kernel_launch(void* const* d_in, const int* in_sizes, int n_in,
                              void* d_out, int out_size, void* d_ws, size_t ws_size,
                              hipStream_t stream) {
    // Inputs in setup_inputs() order:
    const float* x     = (const float*)d_in[0];   // [N*128]
    const int*   ei    = (const int*)  d_in[1];   // [2*E] (src then dst)
    const float* attr  = (const float*)d_in[2];   // [E]
    const int*   etype = (const int*)  d_in[3];   // [E]
    const float* scale = (const float*)d_in[4];   // [4]
    const float* W1    = (const float*)d_in[5];
    const float* b1    = (const float*)d_in[6];
    const float* W2    = (const float*)d_in[7];
    const float* b2    = (const float*)d_in[8];
    const float* W3    = (const float*)d_in[9];
    const float* b3    = (const float*)d_in[10];

    const int E = in_sizes[2];
    const int N = in_sizes[0] / DIM;
    const int* src = ei;
    const int* dst = ei + E;

    // Workspace layout (floats)
    float* ws       = (float*)d_ws;
    float* w_e      = ws;                         // [E]
    float* norm     = w_e + E;                    // [E]
    float* dinv     = norm + E;                   // [N]  (deg -> dinv in place)
    float* selfnorm = dinv + N;                   // [N]
    float* hbuf     = selfnorm + N;               // [N*DIM]  GEMM output
    float* fbuf     = hbuf + (size_t)N * DIM;     // [N*DIM]  aggregated features
    float* out      = (float*)d_out;

    const int TPB = 256;
    const int gN   = (N + TPB - 1) / TPB;
    const int gE   = (E + TPB - 1) / TPB;
    const int gNF  = (N * DIM + TPB - 1) / TPB;
    const int gSc  = (E + 7) / 8;                 // 8 waves per block
    const int gGemm = N / 16;                     // exact: 50000 % 16 == 0

    // Edge normalization (once)
    k_init_deg<<<gN, TPB, 0, stream>>>(dinv, N);
    k_w_deg  <<<gE, TPB, 0, stream>>>(attr, etype, scale, dst, w_e, dinv, E);
    k_dinv   <<<gN, TPB, 0, stream>>>(dinv, selfnorm, N);
    k_norm   <<<gE, TPB, 0, stream>>>(w_e, dinv, src, dst, norm, E);

    // Layer 1: no leaky on input (slope = 1 -> identity)
    k_gemm_wmma<<<gGemm, TPB, 0, stream>>>(x, W1, hbuf, 1.0f);
    k_self_bias<<<gNF, TPB, 0, stream>>>(hbuf, selfnorm, b1, fbuf, N * DIM);
    k_scatter  <<<gSc, TPB, 0, stream>>>(hbuf, norm, src, dst, fbuf, E);

    // Layer 2: leaky fused into A-load
    k_gemm_wmma<<<gGemm, TPB, 0, stream>>>(fbuf, W2, hbuf, NEG_SLOPE);
    k_self_bias<<<gNF, TPB, 0, stream>>>(hbuf, selfnorm, b2, fbuf, N * DIM);
    k_scatter  <<<gSc, TPB, 0, stream>>>(hbuf, norm, src, dst, fbuf, E);

    // Layer 3: final output to d_out (self_bias fully overwrites it first)
    k_gemm_wmma<<<gGemm, TPB, 0, stream>>>(fbuf, W3, hbuf, NEG_SLOPE);
    k_self_bias<<<gNF, TPB, 0, stream>>>(hbuf, selfnorm, b3, out, N * DIM);
    k_scatter  <<<gSc, TPB, 0, stream>>>(hbuf, norm, src, dst, out, E);
}